// Loss_14800457302607
// MI455X (gfx1250) — compile-verified
//
#include <hip/hip_runtime.h>
#include <hip/hip_bf16.h>

#define BATCH 8
#define CHAN 7
#define NPIX (640*640)          /* 409600 */
#define HIST_BINS 2048
#define HIST_BPB 50             /* blocks per batch, hist passes: 8192 px/block */
#define DICE_BPB 50             /* blocks per batch, dice pass:   8192 px/block */
#define TPB 256

typedef __attribute__((ext_vector_type(2))) float v2f;
typedef __attribute__((ext_vector_type(8))) float v8f;

/* workspace byte offsets */
#define OFF_H0   0
#define OFF_H1   65536
#define OFF_H2   131072
#define OFF_CNT  196608
#define OFF_ST   196672
#define OFF_PART 197120

__device__ __forceinline__ unsigned f2key(float f) {
    unsigned u = __float_as_uint(f);
    return (u & 0x80000000u) ? ~u : (u | 0x80000000u);
}
__device__ __forceinline__ float key2f(unsigned k) {
    unsigned u = (k & 0x80000000u) ? (k ^ 0x80000000u) : ~k;
    return __uint_as_float(u);
}
__device__ __forceinline__ float sigf(float x) {
    return 1.0f / (1.0f + __expf(-x));
}

__global__ void init_ws(unsigned* ws, int n) {
    int i = blockIdx.x * blockDim.x + threadIdx.x;
    int stride = gridDim.x * blockDim.x;
    for (; i < n; i += stride) ws[i] = 0u;
}

/* Streaming histogram of monotonic keys of negative text scores.
 * level 0: also counts pos_num / total_neg; bins = key[31:21]
 * level 1: filter key[31:21]==prefix;       bins = key[20:10]
 * level 2: filter key[31:10]==prefix;       bins = key[9:0]
 * All loads are float4 (GLOBAL_LOAD_B128), 4 px per thread-iteration. */
__global__ void hist_pass(const float* __restrict__ outputs,
                          const float* __restrict__ labels,
                          const float* __restrict__ masks,
                          unsigned* __restrict__ hist,
                          unsigned* __restrict__ counts,
                          const unsigned* __restrict__ state,
                          int level) {
    __shared__ unsigned lh[HIST_BINS];
    __shared__ unsigned lcnt[2];
    int b   = blockIdx.x / HIST_BPB;
    int blk = blockIdx.x % HIST_BPB;
    for (int i = threadIdx.x; i < HIST_BINS; i += TPB) lh[i] = 0u;
    if (threadIdx.x < 2) lcnt[threadIdx.x] = 0u;
    unsigned prefix = (level > 0) ? state[b * 4 + 1] : 0u;
    __syncthreads();

    const int chunk = NPIX / HIST_BPB;          /* 8192 */
    const int nv    = chunk / 4;                /* 2048 float4 */
    size_t i0 = (size_t)blk * chunk;
    const float4* s4 = (const float4*)(outputs + ((size_t)b * CHAN + (CHAN - 1)) * NPIX + i0);
    const float4* g4 = (const float4*)(labels  + ((size_t)b * CHAN + (CHAN - 1)) * NPIX + i0);
    const float4* m4 = (const float4*)(masks   + (size_t)b * NPIX + i0);

    unsigned posc = 0, negc = 0;
    for (int v = threadIdx.x; v < nv; v += TPB) {
        __builtin_prefetch(&s4[v + 4 * TPB], 0, 0);
        float4 sv = s4[v];
        float4 gv = g4[v];
        float ss[4] = {sv.x, sv.y, sv.z, sv.w};
        float gg[4] = {gv.x, gv.y, gv.z, gv.w};
        if (level == 0) {
            float4 mv = m4[v];
            float mm[4] = {mv.x, mv.y, mv.z, mv.w};
#pragma unroll
            for (int j = 0; j < 4; j++) {
                bool neg = !(gg[j] > 0.5f);
                if (!neg && mm[j] > 0.5f) posc++;
                if (neg) {
                    negc++;
                    unsigned k = f2key(ss[j]);
                    atomicAdd(&lh[k >> 21], 1u);
                }
            }
        } else if (level == 1) {
#pragma unroll
            for (int j = 0; j < 4; j++) {
                if (!(gg[j] > 0.5f)) {
                    unsigned k = f2key(ss[j]);
                    if ((k >> 21) == prefix) atomicAdd(&lh[(k >> 10) & 0x7FFu], 1u);
                }
            }
        } else {
#pragma unroll
            for (int j = 0; j < 4; j++) {
                if (!(gg[j] > 0.5f)) {
                    unsigned k = f2key(ss[j]);
                    if ((k >> 10) == prefix) atomicAdd(&lh[k & 0x3FFu], 1u);
                }
            }
        }
    }
    if (level == 0) {
        if (posc) atomicAdd(&lcnt[0], posc);
        if (negc) atomicAdd(&lcnt[1], negc);
    }
    __syncthreads();
    unsigned* gh = hist + (size_t)b * HIST_BINS;
    for (int i = threadIdx.x; i < HIST_BINS; i += TPB) {
        unsigned v = lh[i];
        if (v) atomicAdd(&gh[i], v);
    }
    if (level == 0 && threadIdx.x < 2) {
        unsigned v = lcnt[threadIdx.x];
        if (v) atomicAdd(&counts[b * 2 + threadIdx.x], v);
    }
}

/* One block per batch: scan histogram from the top (descending keys),
 * find the bucket containing the k-th largest and the residual rank. */
__global__ void select_pass(const unsigned* __restrict__ hist,
                            const unsigned* __restrict__ counts,
                            unsigned* __restrict__ state,
                            int level, int nbins) {
    int b = blockIdx.x;
    const unsigned* h = hist + (size_t)b * HIST_BINS;
    __shared__ unsigned csum[256];

    unsigned k, skip;
    if (level == 0) {
        unsigned pos  = counts[b * 2 + 0];
        unsigned tneg = counts[b * 2 + 1];
        unsigned long long nn = (unsigned long long)pos * 3ull;
        unsigned neg_num = (nn < (unsigned long long)tneg) ? (unsigned)nn : tneg;
        skip = (pos == 0u || neg_num == 0u) ? 1u : 0u;
        k = neg_num;
    } else {
        k    = state[b * 4 + 0];
        skip = state[b * 4 + 2];
    }

    int CH = nbins / 8;
    if ((int)threadIdx.x < CH) {
        unsigned s = 0;
        for (int j = 0; j < 8; j++) s += h[nbins - 1 - ((int)threadIdx.x * 8 + j)];
        csum[threadIdx.x] = s;
    }
    __syncthreads();

    if (threadIdx.x == 0) {
        unsigned bin0 = 0, rank = 0;
        if (!skip && k > 0) {
            unsigned cum = 0;
            int t = 0;
            for (; t < CH; t++) {
                if (cum + csum[t] >= k) break;
                cum += csum[t];
            }
            if (t == CH) t = CH - 1;
            for (int j = 0; j < 8; j++) {
                int bin = nbins - 1 - (t * 8 + j);
                unsigned c = h[bin];
                if (cum + c >= k) { bin0 = (unsigned)bin; rank = k - cum; break; }
                cum += c;
            }
        }
        if (level == 0) {
            state[b * 4 + 0] = rank;
            state[b * 4 + 1] = bin0;
            state[b * 4 + 2] = skip;
        } else if (level == 1) {
            state[b * 4 + 0] = rank;
            state[b * 4 + 1] = (state[b * 4 + 1] << 11) | bin0;
        } else {
            unsigned key = (state[b * 4 + 1] << 10) | bin0;
            float thr = skip ? -__builtin_inff() : key2f(key);
            state[b * 4 + 3] = __float_as_uint(thr);
        }
    }
}

/* Fused dice pass: text dice (OHEM sel) + 6 kernel channels (sel_k).
 * float4 streaming (GLOBAL_LOAD_B128); per-block combine of the 8
 * wave-partials via v_wmma_f32_16x16x4_f32: D = ones(16x4) x B(4x16) + C,
 * chained on C over wave groups, so every row of D = column sums. */
__global__ void dice_pass(const float* __restrict__ outputs,
                          const float* __restrict__ labels,
                          const float* __restrict__ masks,
                          const unsigned* __restrict__ state,
                          float* __restrict__ partials) {
    int b   = blockIdx.x / DICE_BPB;
    int blk = blockIdx.x % DICE_BPB;
    float thr = __uint_as_float(state[b * 4 + 3]);
    bool  fb  = state[b * 4 + 2] != 0u;

    const int chunk = NPIX / DICE_BPB;          /* 8192 */
    const int nv    = chunk / 4;                /* 2048 float4; 2048/256 = 8 uniform iters */
    size_t i0 = (size_t)blk * chunk;
    const float* base_o = outputs + (size_t)b * CHAN * NPIX + i0;
    const float* base_l = labels  + (size_t)b * CHAN * NPIX + i0;
    const float4* m4 = (const float4*)(masks + (size_t)b * NPIX + i0);
    const float4* s4 = (const float4*)(base_o + (size_t)(CHAN - 1) * NPIX);
    const float4* g4 = (const float4*)(base_l + (size_t)(CHAN - 1) * NPIX);

    float accA[7], accB[7], accC[7];
#pragma unroll
    for (int c = 0; c < 7; c++) { accA[c] = 0.f; accB[c] = 0.f; accC[c] = 0.f; }

    for (int v = threadIdx.x; v < nv; v += TPB) {
        __builtin_prefetch(&s4[v + 4 * TPB], 0, 0);
        float4 mv = m4[v];
        float4 sv = s4[v];
        float4 gv = g4[v];
        float mm[4] = {mv.x, mv.y, mv.z, mv.w};
        float ss[4] = {sv.x, sv.y, sv.z, sv.w};
        float gg[4] = {gv.x, gv.y, gv.z, gv.w};
        float selK[4];
#pragma unroll
        for (int j = 0; j < 4; j++) {
            bool mok = mm[j] > 0.5f;
            float selT = fb ? mm[j]
                            : ((((ss[j] >= thr) || (gg[j] > 0.5f)) && mok) ? 1.f : 0.f);
            float p = selT * sigf(ss[j]);
            float t = gg[j] * selT;
            accA[6] += p * t; accB[6] += p * p; accC[6] += t * t;
            selK[j] = ((ss[j] > 0.f) && mok) ? 1.f : 0.f;
        }
#pragma unroll
        for (int ch = 0; ch < 6; ch++) {
            const float4* k4 = (const float4*)(base_o + (size_t)ch * NPIX);
            const float4* t4 = (const float4*)(base_l + (size_t)ch * NPIX);
            float4 kv = k4[v];
            float4 tv = t4[v];
            float kk[4] = {kv.x, kv.y, kv.z, kv.w};
            float tt[4] = {tv.x, tv.y, tv.z, tv.w};
#pragma unroll
            for (int j = 0; j < 4; j++) {
                float pk = selK[j] * sigf(kk[j]);
                float tk = tt[j] * selK[j];
                accA[ch] += pk * tk; accB[ch] += pk * pk; accC[ch] += tk * tk;
            }
        }
    }

    /* intra-wave reduce (wave32), partials to LDS as part[q][wave] */
    __shared__ float part[32][8];
    int lane = threadIdx.x & 31;
    int wid  = threadIdx.x >> 5;   /* 0..7 */
#pragma unroll
    for (int ch = 0; ch < 7; ch++) {
        float va = accA[ch], vb = accB[ch], vc = accC[ch];
#pragma unroll
        for (int off = 16; off > 0; off >>= 1) {
            va += __shfl_xor(va, off, 32);
            vb += __shfl_xor(vb, off, 32);
            vc += __shfl_xor(vc, off, 32);
        }
        if (lane == 0) {
            part[ch * 3 + 0][wid] = va;
            part[ch * 3 + 1][wid] = vb;
            part[ch * 3 + 2][wid] = vc;
        }
    }
    if (lane == 0) {
        for (int q = 21; q < 32; q++) part[q][wid] = 0.f;
    }
    __syncthreads();

#if __has_builtin(__builtin_amdgcn_wmma_f32_16x16x4_f32)
    /* All waves execute redundantly so EXEC is all-ones; only wave 0 stores.
     * Column n of B holds 4 wave-partials of quantity (grp*16+n); A = ones,
     * so D[m][n] = sum_k B[k][n] + C[m][n], chained over wave groups. */
    {
        v2f ones; ones.x = 1.f; ones.y = 1.f;
        int col = lane & 15, hi = lane >> 4;
#pragma unroll
        for (int grp = 0; grp < 2; grp++) {
            v8f c = {0.f, 0.f, 0.f, 0.f, 0.f, 0.f, 0.f, 0.f};
#pragma unroll
            for (int w0 = 0; w0 < 8; w0 += 4) {
                v2f bm;
                bm.x = part[grp * 16 + col][w0 + 2 * hi + 0];
                bm.y = part[grp * 16 + col][w0 + 2 * hi + 1];
                c = __builtin_amdgcn_wmma_f32_16x16x4_f32(
                        false, ones, false, bm, (short)0, c, false, false);
            }
            int q = grp * 16 + col;
            if (wid == 0 && hi == 0 && q < 21)
                partials[(size_t)blockIdx.x * 32 + q] = c[0];
        }
    }
#else
    if (wid == 0 && lane < 21) {
        float s = 0.f;
        for (int w = 0; w < 8; w++) s += part[lane][w];
        partials[(size_t)blockIdx.x * 32 + lane] = s;
    }
#endif
}

__global__ void final_reduce(const float* __restrict__ partials,
                             float* __restrict__ out) {
    __shared__ float sums[BATCH * 21];
    int tid = threadIdx.x;
    if (tid < BATCH * 21) {
        int b = tid / 21, q = tid % 21;
        float s = 0.f;
        for (int j = 0; j < DICE_BPB; j++)
            s += partials[(size_t)(b * DICE_BPB + j) * 32 + q];
        sums[tid] = s;
    }
    __syncthreads();
    if (tid == 0) {
        float lt = 0.f, lk = 0.f;
        for (int b = 0; b < BATCH; b++) {
            const float* sb = &sums[b * 21];
            float a  = sb[6 * 3 + 0];
            float bb = sb[6 * 3 + 1] + 0.001f;
            float cc = sb[6 * 3 + 2] + 0.001f;
            lt += 1.f - 2.f * a / (bb + cc);
            float lkb = 0.f;
            for (int ch = 0; ch < 6; ch++) {
                float ka  = sb[ch * 3 + 0];
                float kb_ = sb[ch * 3 + 1] + 0.001f;
                float kc  = sb[ch * 3 + 2] + 0.001f;
                lkb += 1.f - 2.f * ka / (kb_ + kc);
            }
            lk += lkb * (1.f / 6.f);
        }
        lt *= (1.f / BATCH);
        lk *= (1.f / BATCH);
        out[0] = lt;
        out[1] = lk;
        out[2] = 0.7f * lt + 0.3f * lk;
    }
}

extern "C" void kernel_launch(void* const* d_in, const int* in_sizes, int n_in,
                              void* d_out, int out_size, void* d_ws, size_t ws_size,
                              hipStream_t stream) {
    (void)in_sizes; (void)n_in; (void)out_size; (void)ws_size;
    const float* outputs = (const float*)d_in[0];
    const float* labels  = (const float*)d_in[1];
    const float* masks   = (const float*)d_in[2];
    float* out = (float*)d_out;
    char* ws = (char*)d_ws;

    unsigned* h0     = (unsigned*)(ws + OFF_H0);
    unsigned* h1     = (unsigned*)(ws + OFF_H1);
    unsigned* h2     = (unsigned*)(ws + OFF_H2);
    unsigned* counts = (unsigned*)(ws + OFF_CNT);
    unsigned* state  = (unsigned*)(ws + OFF_ST);
    float* partials  = (float*)(ws + OFF_PART);

    init_ws<<<64, 256, 0, stream>>>((unsigned*)ws, OFF_PART / 4);

    hist_pass<<<BATCH * HIST_BPB, TPB, 0, stream>>>(outputs, labels, masks, h0, counts, state, 0);
    select_pass<<<BATCH, 256, 0, stream>>>(h0, counts, state, 0, 2048);
    hist_pass<<<BATCH * HIST_BPB, TPB, 0, stream>>>(outputs, labels, masks, h1, counts, state, 1);
    select_pass<<<BATCH, 256, 0, stream>>>(h1, counts, state, 1, 2048);
    hist_pass<<<BATCH * HIST_BPB, TPB, 0, stream>>>(outputs, labels, masks, h2, counts, state, 2);
    select_pass<<<BATCH, 256, 0, stream>>>(h2, counts, state, 2, 1024);

    dice_pass<<<BATCH * DICE_BPB, TPB, 0, stream>>>(outputs, labels, masks, state, partials);
    final_reduce<<<1, 256, 0, stream>>>(partials, out);
}